// SparseVAE2_34376918237635
// MI455X (gfx1250) — compile-verified
//
#include <hip/hip_runtime.h>
#include <hip/hip_bf16.h>
#include <math.h>

typedef float v2f __attribute__((ext_vector_type(2)));
typedef float v8f __attribute__((ext_vector_type(8)));

// ---------------------------------------------------------------------------
// Implicit-GEMM conv / transposed-conv with V_WMMA_F32_16X16X4_F32 (wave32).
//
// Unified gather rule (lax.conv is correlation; lax.conv_transpose with
// transpose_kernel=False is correlation over the lhs-dilated input with
// pad_lo = k-1):  t = o + k - pad; valid iff t>=0 && t%s==0 && t/s < In.
// Forward conv is the s==1 special case.
//
// GEMM mapping (per batch image):
//   M = Hout*Wout output positions (16-tile), N = Cout (16-tile),
//   K ordered tap-major: k = (kh*KW+kw)*CinPad4 + ci, so each kernel tap
//   resolves its spatial gather ONCE and the inner ci loop is pure
//   strided loads + WMMA.
//
// WMMA f32 16x16x4 VGPR layout (ISA 7.12.2):
//   A: lane L -> M=L&15, VGPR0/1 = K = 2*(L>=16) + {0,1}
//   B: VGPR i -> row K = i + 2*(L>=16), N = L&15
//   D: VGPR i -> M = i + 8*(L>=16),     N = L&15
// ---------------------------------------------------------------------------

template <int Cin, int KH, int KW, int padH, int padW, int sH, int sW, int LEAKY>
__global__ void __launch_bounds__(256)
conv_wmma(const float* __restrict__ in, const float* __restrict__ wgt,
          float* __restrict__ out,
          int Hin, int Win, int Cout, int Hout, int Wout,
          int nT, int tilesPerB, int totalWaves)
{
    const int lane = threadIdx.x & 31;
    const int wave = blockIdx.x * (blockDim.x >> 5) + (threadIdx.x >> 5);
    if (wave >= totalWaves) return;

    const int b  = wave / tilesPerB;
    const int t  = wave - b * tilesPerB;
    const int nt = t % nT;
    const int mt = t / nT;

    const int half = lane >> 4;     // K sub-block: lanes 0-15 -> K{0,1}, 16-31 -> K{2,3}
    const int l15  = lane & 15;
    const int M    = Hout * Wout;
    constexpr int khkw = KH * KW;

    // A-side: output position of this lane's GEMM row
    const int  mRow   = mt * 16 + l15;
    const bool mValid = mRow < M;
    int oy = 0, ox = 0;
    if (mValid) { oy = mRow / Wout; ox = mRow - oy * Wout; }

    // B-side: output channel of this lane's GEMM column
    const int  nCol   = nt * 16 + l15;
    const bool nValid = nCol < Cout;
    const float* __restrict__ wN =
        wgt + (size_t)(nValid ? nCol : 0) * Cin * khkw;   // w[n][ci][kh][kw]
    __builtin_prefetch(wN, 0, 1);                          // global_prefetch_b8

    const float* __restrict__ inB = in + (size_t)b * Cin * Hin * Win;
    const int planeStride = Hin * Win;                     // A stride per ci
    const int ci0 = half * 2;

    v8f acc = {0.f, 0.f, 0.f, 0.f, 0.f, 0.f, 0.f, 0.f};

#pragma unroll
    for (int kh = 0; kh < KH; ++kh) {
#pragma unroll
        for (int kw = 0; kw < KW; ++kw) {
            // Resolve this tap's input coords once (constant divisors).
            const int ty = oy + kh - padH;
            const int tx = ox + kw - padW;
            bool ok = mValid && ty >= 0 && tx >= 0;
            int iy = ty, ix = tx;
            if (sH != 1) { ok = ok && (ty % sH) == 0; iy = ty / sH; }
            if (sW != 1) { ok = ok && (tx % sW) == 0; ix = tx / sW; }
            ok = ok && iy < Hin && ix < Win;
            const float* __restrict__ aPtr =
                inB + (size_t)(ok ? (iy * Win + ix) : 0);
            const float* __restrict__ bPtr = wN + kh * KW + kw;

            // Inner reduction over input channels, 4 per WMMA.
#pragma unroll
            for (int ci = 0; ci < Cin; ci += 4) {
                const int c0 = ci + ci0;
                const int c1 = c0 + 1;
                v2f av, bv;
                av[0] = (ok && c0 < Cin) ? aPtr[(size_t)c0 * planeStride] : 0.0f;
                av[1] = (ok && c1 < Cin) ? aPtr[(size_t)c1 * planeStride] : 0.0f;
                bv[0] = (nValid && c0 < Cin) ? bPtr[(size_t)c0 * khkw] : 0.0f;
                bv[1] = (nValid && c1 < Cin) ? bPtr[(size_t)c1 * khkw] : 0.0f;
                acc = __builtin_amdgcn_wmma_f32_16x16x4_f32(
                    /*neg_a=*/false, av, /*neg_b=*/false, bv,
                    /*c_mod=*/(short)0, acc, /*reuse_a=*/false, /*reuse_b=*/false);
            }
        }
    }

    // D 16x16 f32: VGPR i -> M = i + 8*half, N = l15
    if (nValid) {
        float* __restrict__ orow = out + ((size_t)b * Cout + nCol) * M;
#pragma unroll
        for (int i = 0; i < 8; ++i) {
            const int m = mt * 16 + i + half * 8;
            if (m < M) {
                float v = acc[i];
                if (LEAKY) v = (v >= 0.0f) ? v : 0.01f * v;
                orow[m] = v;
            }
        }
    }
}

// ---------------------------------------------------------------------------
// Non-overlapping max pool (window == stride), one thread per output element.
// ---------------------------------------------------------------------------
__global__ void __launch_bounds__(256)
maxpool_k(const float* __restrict__ in, float* __restrict__ out,
          int total, int Hin, int Win, int ph, int pw, int Hout, int Wout)
{
    int i = blockIdx.x * blockDim.x + threadIdx.x;
    if (i >= total) return;
    int ox = i % Wout;
    int t  = i / Wout;
    int oy = t % Hout;
    int bc = t / Hout;
    const float* p = in + ((size_t)bc * Hin + (size_t)oy * ph) * Win + (size_t)ox * pw;
    float m = -INFINITY;
    for (int dy = 0; dy < ph; ++dy)
        for (int dx = 0; dx < pw; ++dx)
            m = fmaxf(m, p[(size_t)dy * Win + dx]);
    out[i] = m;
}

// ---------------------------------------------------------------------------
// Reparameterization: z = mu + eps * exp(0.5*logvar); deterministic eps via
// PCG hash + Box-Muller. Also emits mu/logvar/z into the output tail.
// ---------------------------------------------------------------------------
__device__ __forceinline__ unsigned pcg_hash(unsigned x)
{
    x = x * 747796405u + 2891336453u;
    x = ((x >> ((x >> 28) + 4u)) ^ x) * 277803737u;
    return (x >> 22) ^ x;
}

__global__ void __launch_bounds__(256)
reparam_k(const float* __restrict__ mu, const float* __restrict__ lv,
          float* __restrict__ z,
          float* __restrict__ omu, float* __restrict__ olv,
          float* __restrict__ oz, int n)
{
    int i = blockIdx.x * blockDim.x + threadIdx.x;
    if (i >= n) return;
    unsigned h1 = pcg_hash(0x9E3779B9u ^ (unsigned)(2 * i));
    unsigned h2 = pcg_hash(0x85EBCA6Bu ^ (unsigned)(2 * i + 1));
    float u1 = ((float)h1 + 1.0f) * (1.0f / 4294967296.0f);   // (0,1]
    float u2 = (float)h2 * (1.0f / 4294967296.0f);
    float eps = sqrtf(fmaxf(-2.0f * __logf(u1), 0.0f)) * __cosf(6.28318530718f * u2);
    float m = mu[i], l = lv[i];
    float zz = m + eps * __expf(0.5f * l);
    z[i] = zz;
    omu[i] = m; olv[i] = l; oz[i] = zz;
}

// ---------------------------------------------------------------------------
// Host-side launch helpers
// ---------------------------------------------------------------------------
template <int Cin, int KH, int KW, int pH, int pW, int sH, int sW, int LK>
static void launch_conv(const float* in, const float* w, float* out,
                        int B, int Hin, int Win,
                        int Cout, int Hout, int Wout, hipStream_t s)
{
    const int M  = Hout * Wout;
    const int mT = (M + 15) / 16;
    const int nT = (Cout + 15) / 16;
    const int tilesPerB  = mT * nT;
    const int totalWaves = B * tilesPerB;
    const int blocks = (totalWaves + 7) / 8;   // 256 threads = 8 wave32
    conv_wmma<Cin, KH, KW, pH, pW, sH, sW, LK>
        <<<dim3(blocks), dim3(256), 0, s>>>(
            in, w, out, Hin, Win, Cout, Hout, Wout, nT, tilesPerB, totalWaves);
}

static void launch_pool(const float* in, float* out, int BC, int Hin, int Win,
                        int ph, int pw, hipStream_t s)
{
    int Hout = Hin / ph, Wout = Win / pw;
    int total = BC * Hout * Wout;
    maxpool_k<<<dim3((total + 255) / 256), dim3(256), 0, s>>>(
        in, out, total, Hin, Win, ph, pw, Hout, Wout);
}

extern "C" void kernel_launch(void* const* d_in, const int* in_sizes, int n_in,
                              void* d_out, int out_size, void* d_ws, size_t ws_size,
                              hipStream_t stream)
{
    (void)in_sizes; (void)n_in; (void)out_size; (void)ws_size;

    const float* x   = (const float*)d_in[0];
    const float* w1  = (const float*)d_in[1];   // (8,6,3,3)
    const float* w2  = (const float*)d_in[2];   // (16,8,3,3)
    const float* w3  = (const float*)d_in[3];   // (32,16,3,3)
    const float* wmu = (const float*)d_in[4];   // (1,32,1,5)
    const float* wlv = (const float*)d_in[5];   // (1,32,1,5)
    const float* wu0 = (const float*)d_in[6];   // (32,1,1,5)
    const float* wu1 = (const float*)d_in[7];   // (32,32,5,2)
    const float* w4  = (const float*)d_in[8];   // (16,32,3,3)
    const float* wu2 = (const float*)d_in[9];   // (16,16,5,2)
    const float* w5  = (const float*)d_in[10];  // (8,16,3,3)
    const float* wu3 = (const float*)d_in[11];  // (8,8,2,2)
    const float* w6  = (const float*)d_in[12];  // (6,8,3,3)

    float* out = (float*)d_out;
    float* ws  = (float*)d_ws;

    // Ping-pong workspace (floats): A=16.384M, B=16.384M, + latent slices.
    float* A  = ws;
    float* Bf = ws + 16384000;
    float* MU = ws + 32768000;
    float* LV = MU + 3840;
    float* Z  = LV + 3840;

    const int B = 32;

    // ---------------- Encoder ----------------
    // conv1 6->8, 3x3 SAME, lrelu : [32,8,500,128]
    launch_conv<6, 3, 3, 1, 1, 1, 1, 1>(x,  w1, A,  B, 500, 128,  8, 500, 128, stream);
    // maxpool 2x2 -> [32,8,250,64]
    launch_pool(A, Bf, B * 8, 500, 128, 2, 2, stream);
    // conv2 8->16, lrelu : [32,16,250,64]
    launch_conv<8, 3, 3, 1, 1, 1, 1, 1>(Bf, w2, A,  B, 250, 64, 16, 250, 64, stream);
    // maxpool 5x2 -> [32,16,50,32]
    launch_pool(A, Bf, B * 16, 250, 64, 5, 2, stream);
    // conv3 16->32, lrelu : [32,32,50,32]
    launch_conv<16, 3, 3, 1, 1, 1, 1, 1>(Bf, w3, A,  B, 50, 32, 32, 50, 32, stream);
    // maxpool 5x2 -> [32,32,10,16]
    launch_pool(A, Bf, B * 32, 50, 32, 5, 2, stream);
    // mu / logvar : 32->1, kernel (1,5) VALID -> [32,1,10,12]
    launch_conv<32, 1, 5, 0, 0, 1, 1, 0>(Bf, wmu, MU, B, 10, 16, 1, 10, 12, stream);
    launch_conv<32, 1, 5, 0, 0, 1, 1, 0>(Bf, wlv, LV, B, 10, 16, 1, 10, 12, stream);

    // ---------------- Reparameterize ----------------
    reparam_k<<<dim3((3840 + 255) / 256), dim3(256), 0, stream>>>(
        MU, LV, Z, out + 12288000, out + 12291840, out + 12295680, 3840);

    // ---------------- Decoder (convT: pad = k-1, stride = lhs dilation) ----
    // convT wu0 1->32, k(1,5) s(1,1) : [32,32,10,16]
    launch_conv<1, 1, 5, 0, 4, 1, 1, 1>(Z,  wu0, A,  B, 10, 12, 32, 10, 16, stream);
    // convT wu1 32->32, k(5,2) s(5,2) : [32,32,50,32]
    launch_conv<32, 5, 2, 4, 1, 5, 2, 1>(A,  wu1, Bf, B, 10, 16, 32, 50, 32, stream);
    // conv w4 32->16 SAME : [32,16,50,32]
    launch_conv<32, 3, 3, 1, 1, 1, 1, 1>(Bf, w4,  A,  B, 50, 32, 16, 50, 32, stream);
    // convT wu2 16->16, k(5,2) s(5,2) : [32,16,250,64]
    launch_conv<16, 5, 2, 4, 1, 5, 2, 1>(A,  wu2, Bf, B, 50, 32, 16, 250, 64, stream);
    // conv w5 16->8 SAME : [32,8,250,64]
    launch_conv<16, 3, 3, 1, 1, 1, 1, 1>(Bf, w5,  A,  B, 250, 64, 8, 250, 64, stream);
    // convT wu3 8->8, k(2,2) s(2,2) : [32,8,500,128]
    launch_conv<8, 2, 2, 1, 1, 2, 2, 1>(A,  wu3, Bf, B, 250, 64, 8, 500, 128, stream);
    // conv w6 8->6 SAME, lrelu -> d_out head : [32,6,500,128]
    launch_conv<8, 3, 3, 1, 1, 1, 1, 1>(Bf, w6, out, B, 500, 128, 6, 500, 128, stream);
}